// GCN_51049981281479
// MI455X (gfx1250) — compile-verified
//
#include <hip/hip_runtime.h>
#include <hip/hip_bf16.h>

// ---------------------------------------------------------------------------
// GCN 2-layer forward for MI455X (gfx1250, wave32).
//   layer: mean-aggregate over edges (L2-resident f32 atomics), then dense
//   fp32 GEMM on the matrix pipe via V_WMMA_F32_16X16X4_F32.
// ---------------------------------------------------------------------------

typedef float v2f __attribute__((ext_vector_type(2)));
typedef float v8f __attribute__((ext_vector_type(8)));

#define IN_FEATS 64
#define HIDDEN 64
#define NUM_CLASSES 16

// ---------------------------------------------------------------------------
// Zero a float buffer (grid-stride, float4 wide). n must be a multiple of 4
// in the fast region; scalar tail handled explicitly.
// ---------------------------------------------------------------------------
__global__ void gcn_zero_kernel(float* __restrict__ p, long long n) {
    long long i = (long long)blockIdx.x * blockDim.x + threadIdx.x;
    long long stride = (long long)gridDim.x * blockDim.x;
    long long n4 = n >> 2;
    float4* __restrict__ p4 = (float4*)p;
    for (long long j = i; j < n4; j += stride)
        p4[j] = make_float4(0.f, 0.f, 0.f, 0.f);
    // scalar tail (n not multiple of 4)
    long long tail = n4 << 2;
    for (long long j = tail + i; j < n; j += stride) p[j] = 0.0f;
}

// ---------------------------------------------------------------------------
// Edge scatter: sum[dst] += x[src] (64 floats / edge, 16 threads x float4),
// optionally deg[dst] += 1.  Accumulator (25.6MB) + edges (25.6MB) + source
// rows (25.6MB) all fit in the 192MB L2, so atomics resolve at L2.
// ---------------------------------------------------------------------------
__global__ void gcn_scatter_kernel(const float* __restrict__ x,
                                   const long long* __restrict__ src,
                                   const long long* __restrict__ dst,
                                   float* __restrict__ sum,
                                   float* __restrict__ deg,   // may be null
                                   int n_edges) {
    int t = blockIdx.x * blockDim.x + threadIdx.x;
    int edge = t >> 4;        // 16 threads per edge
    int chunk = t & 15;       // 4 floats per thread
    if (edge >= n_edges) return;
    long long s = src[edge];
    long long d = dst[edge];
    const float4 v = *(const float4*)(x + s * 64 + chunk * 4);
    float* o = sum + d * 64 + chunk * 4;
    atomicAdd(o + 0, v.x);
    atomicAdd(o + 1, v.y);
    atomicAdd(o + 2, v.z);
    atomicAdd(o + 3, v.w);
    if (deg != nullptr && chunk == 0) atomicAdd(deg + d, 1.0f);
}

// ---------------------------------------------------------------------------
// Layer 1: h = relu( (sum/deg) @ W1 + b1 ), W1 is 64x64.
// One wave per 16-row node tile; 4 N-tiles x 16 K-steps of v_wmma_f32_16x16x4_f32.
// A fragments (one row per lane, mean-divide fused) hoisted across N-tiles.
// Stores: wave-uniform fast path (no per-element exec masking) for full tiles.
// ---------------------------------------------------------------------------
__global__ void gcn_gemm1_relu_kernel(const float* __restrict__ sum,
                                      const float* __restrict__ deg,
                                      const float* __restrict__ W1,
                                      const float* __restrict__ b1,
                                      float* __restrict__ h,
                                      int n_nodes) {
    const int lane = threadIdx.x & 31;
    const int wave = threadIdx.x >> 5;
    const int tile = blockIdx.x * (blockDim.x >> 5) + wave;
    const int row0 = tile * 16;
    if (row0 >= n_nodes) return;                  // wave-uniform exit

    const int m    = lane & 15;                   // col (B/C/D) & row (A) in tile
    const int koff = (lane >> 4) << 1;            // A/B K offset: 0 or 2
    const int mhi  = (lane >> 4) << 3;            // C/D row offset: 0 or 8
    const bool fullTile = (row0 + 16 <= n_nodes); // wave-uniform

    // A fragment: row = row0+m (clamped for tail tile), mean fused here.
    int row = row0 + m;
    int rowc = (row < n_nodes) ? row : (n_nodes - 1);
    const float invd = 1.0f / fmaxf(deg[rowc], 1.0f);
    const float* __restrict__ arow = sum + (long long)rowc * IN_FEATS;

    v2f afrag[16];
#pragma unroll
    for (int kk = 0; kk < 16; ++kk) {
        afrag[kk].x = arow[kk * 4 + koff + 0] * invd;
        afrag[kk].y = arow[kk * 4 + koff + 1] * invd;
    }

#pragma unroll
    for (int nt = 0; nt < 4; ++nt) {
        const int n0 = nt * 16;
        v8f c = {};
#pragma unroll
        for (int kk = 0; kk < 16; ++kk) {
            const int k = kk * 4 + koff;
            v2f b;
            b.x = W1[(k + 0) * HIDDEN + n0 + m];
            b.y = W1[(k + 1) * HIDDEN + n0 + m];
            c = __builtin_amdgcn_wmma_f32_16x16x4_f32(
                    false, afrag[kk], false, b, (short)0, c, false, false);
        }
        const float bias = b1[n0 + m];
        if (fullTile) {
            // full 16-row tile: unconditional, fully coalesced stores
#pragma unroll
            for (int v = 0; v < 8; ++v) {
                const int r = row0 + v + mhi;     // C/D layout: M = v + (lane<16?0:8)
                h[(long long)r * HIDDEN + n0 + m] = fmaxf(c[v] + bias, 0.0f);
            }
        } else {
#pragma unroll
            for (int v = 0; v < 8; ++v) {
                const int r = row0 + v + mhi;
                if (r < n_nodes)
                    h[(long long)r * HIDDEN + n0 + m] = fmaxf(c[v] + bias, 0.0f);
            }
        }
    }
}

// ---------------------------------------------------------------------------
// Layer 2: out = (sum/deg) @ W2 + b2, W2 is 64x16. One 16x16 tile per wave.
// ---------------------------------------------------------------------------
__global__ void gcn_gemm2_kernel(const float* __restrict__ sum,
                                 const float* __restrict__ deg,
                                 const float* __restrict__ W2,
                                 const float* __restrict__ b2,
                                 float* __restrict__ out,
                                 int n_nodes) {
    const int lane = threadIdx.x & 31;
    const int wave = threadIdx.x >> 5;
    const int tile = blockIdx.x * (blockDim.x >> 5) + wave;
    const int row0 = tile * 16;
    if (row0 >= n_nodes) return;                  // wave-uniform exit

    const int m    = lane & 15;
    const int koff = (lane >> 4) << 1;
    const int mhi  = (lane >> 4) << 3;
    const bool fullTile = (row0 + 16 <= n_nodes); // wave-uniform

    int row = row0 + m;
    int rowc = (row < n_nodes) ? row : (n_nodes - 1);
    const float invd = 1.0f / fmaxf(deg[rowc], 1.0f);
    const float* __restrict__ arow = sum + (long long)rowc * HIDDEN;

    v8f c = {};
#pragma unroll
    for (int kk = 0; kk < 16; ++kk) {
        const int k = kk * 4 + koff;
        v2f a, b;
        a.x = arow[k + 0] * invd;
        a.y = arow[k + 1] * invd;
        b.x = W2[(k + 0) * NUM_CLASSES + m];
        b.y = W2[(k + 1) * NUM_CLASSES + m];
        c = __builtin_amdgcn_wmma_f32_16x16x4_f32(
                false, a, false, b, (short)0, c, false, false);
    }
    const float bias = b2[m];
    if (fullTile) {
#pragma unroll
        for (int v = 0; v < 8; ++v) {
            const int r = row0 + v + mhi;
            out[(long long)r * NUM_CLASSES + m] = c[v] + bias;
        }
    } else {
#pragma unroll
        for (int v = 0; v < 8; ++v) {
            const int r = row0 + v + mhi;
            if (r < n_nodes)
                out[(long long)r * NUM_CLASSES + m] = c[v] + bias;
        }
    }
}

// ---------------------------------------------------------------------------
// Host-side orchestration (all kernels on `stream`; graph-capture safe).
// Workspace layout: [sum: N*64 f32][deg: N(padded) f32][h: N*64 f32] ~52MB.
// ---------------------------------------------------------------------------
extern "C" void kernel_launch(void* const* d_in, const int* in_sizes, int n_in,
                              void* d_out, int out_size, void* d_ws, size_t ws_size,
                              hipStream_t stream) {
    const float*     features = (const float*)d_in[0];
    const long long* src      = (const long long*)d_in[1];
    const long long* dst      = (const long long*)d_in[2];
    const float*     W1       = (const float*)d_in[3];
    const float*     b1       = (const float*)d_in[4];
    const float*     W2       = (const float*)d_in[5];
    const float*     b2       = (const float*)d_in[6];
    float*           out      = (float*)d_out;

    const int n_nodes = in_sizes[0] / IN_FEATS;
    const int n_edges = in_sizes[1];

    float* sum = (float*)d_ws;
    const long long sumElems = (long long)n_nodes * IN_FEATS;
    float* deg = sum + sumElems;
    const long long degPad = ((long long)n_nodes + 3) & ~3LL;   // keep h 16B aligned
    float* h = deg + degPad;

    const int ZB = 256;
    const int zeroGrid = 2048;

    const int SB = 256;
    const long long scatterWork = (long long)n_edges * 16;
    const int scatterGrid = (int)((scatterWork + SB - 1) / SB);

    const int GB = 128;                           // 4 waves / block
    const int tiles = (n_nodes + 15) / 16;
    const int gemmGrid = (tiles + (GB / 32) - 1) / (GB / 32);

    // ---- Layer 1 ----
    gcn_zero_kernel<<<zeroGrid, ZB, 0, stream>>>(sum, sumElems + degPad);
    gcn_scatter_kernel<<<scatterGrid, SB, 0, stream>>>(features, src, dst,
                                                       sum, deg, n_edges);
    gcn_gemm1_relu_kernel<<<gemmGrid, GB, 0, stream>>>(sum, deg, W1, b1,
                                                       h, n_nodes);

    // ---- Layer 2 (degree is identical; only re-zero the accumulator) ----
    gcn_zero_kernel<<<zeroGrid, ZB, 0, stream>>>(sum, sumElems);
    gcn_scatter_kernel<<<scatterGrid, SB, 0, stream>>>(h, src, dst,
                                                       sum, nullptr, n_edges);
    gcn_gemm2_kernel<<<gemmGrid, GB, 0, stream>>>(sum, deg, W2, b2,
                                                  out, n_nodes);
}